// CliquesOutputLayer_75831942578504
// MI455X (gfx1250) — compile-verified
//
#include <hip/hip_runtime.h>
#include <math.h>

// Problem constants (from the reference)
#define Bsz      4
#define NATOMS   100000
#define Dd       64
#define Ff       8
#define Gg       20000
#define Aa       3
#define Hh       16
#define Kk       (Aa * Dd)          // 192
#define TILES_BF (Gg / 16)          // 1250 tiles of 16 groundings per (b,f)
#define CHUNKS   40                 // grid.x: G-chunks per (b,f)
#define WAVES    8
#define THREADS  (WAVES * 32)

typedef __attribute__((ext_vector_type(2))) float v2f;
typedef __attribute__((ext_vector_type(8))) float v8f;

__device__ __forceinline__ float sigmoidf_(float x) {
    return 1.0f / (1.0f + __expf(-x));
}

__global__ __launch_bounds__(THREADS)
void cliques_mlp_wmma(const float* __restrict__ atoms,   // [B, NATOMS, D]
                      const int*   __restrict__ gidx,    // [F, G, A]
                      const float* __restrict__ W1,      // [F, K, H]
                      const float* __restrict__ b1,      // [F, H]
                      const float* __restrict__ W2,      // [F, H, 1]
                      const float* __restrict__ b2,      // [F, 1]
                      float*       __restrict__ out)     // [B, F*G]
{
    __shared__ float sW1[Kk * Hh];   // 12 KB, per-formula weights
    __shared__ float sB1[Hh];
    __shared__ float sW2[Hh];
    __shared__ float sB2[1];

    const int chunk = blockIdx.x;
    const int f     = blockIdx.y;
    const int b     = blockIdx.z;
    const int tid   = threadIdx.x;
    const int wave  = tid >> 5;
    const int lane  = tid & 31;
    const int row   = lane & 15;   // A-row / B-col within tile
    const int half  = lane >> 4;   // K-pair selector (A layout) / C-row half

    // Stage W1[f], b1[f], W2[f], b2[f] into LDS once per block.
    for (int i = tid; i < Kk * Hh; i += THREADS)
        sW1[i] = W1[(size_t)f * (Kk * Hh) + i];
    if (tid < Hh) {
        sB1[tid] = b1[f * Hh + tid];
        sW2[tid] = W2[f * Hh + tid];
    }
    if (tid == 0) sB2[0] = b2[f];
    __syncthreads();

    const float bb1 = sB1[row];   // bias for hidden unit n = row (this lane's column)
    const float w2r = sW2[row];   // W2 weight for hidden unit n = row
    const float bb2 = sB2[0];

    const float* atomsB = atoms + (size_t)b * NATOMS * Dd;
    float* outB = out + ((size_t)b * Ff + f) * Gg;

    // Tile loop: wave-uniform trip count -> EXEC all-ones at every WMMA.
    for (int t = chunk * WAVES + wave; t < TILES_BF; t += CHUNKS * WAVES) {
        const int g = t * 16 + row;
        const int* gi = gidx + ((size_t)f * Gg + g) * Aa;
        const float* arow0 = atomsB + (size_t)gi[0] * Dd;
        const float* arow1 = atomsB + (size_t)gi[1] * Dd;
        const float* arow2 = atomsB + (size_t)gi[2] * Dd;
        const float* arow[Aa] = { arow0, arow1, arow2 };

        v8f c = {};   // fp32 accumulators: rows half*8+0..7, col = row

        #pragma unroll
        for (int a = 0; a < Aa; ++a) {
            const float* ap = arow[a];
            const float* wp = &sW1[a * Dd * Hh];
            #pragma unroll
            for (int s = 0; s < 16; ++s) {
                // K-block base = a*64 + 4*s ; this lane supplies K-pair (kl, kl+1)
                const int kl = 4 * s + 2 * half;

                // A fragment (16x4 f32): lane row M=row, two consecutive K values.
                const float2 av2 = *reinterpret_cast<const float2*>(ap + kl);
                v2f av; av.x = av2.x; av.y = av2.y;

                // B fragment (4x16 f32): col N=row, same K-pair, from LDS.
                v2f bv;
                bv.x = wp[(size_t)kl * Hh + row];
                bv.y = wp[(size_t)(kl + 1) * Hh + row];

                c = __builtin_amdgcn_wmma_f32_16x16x4_f32(
                        /*neg_a=*/false, av,
                        /*neg_b=*/false, bv,
                        /*c_mod=*/(short)0, c,
                        /*reuse_a=*/false, /*reuse_b=*/false);
            }
        }

        // Layer 1 activation + layer-2 partial product (this lane's hidden unit).
        float vals[8];
        #pragma unroll
        for (int j = 0; j < 8; ++j) {
            const float h = sigmoidf_(c[j] + bb1);
            vals[j] = h * w2r;
        }

        // Reduce over the 16 hidden units: xor masks 1,2,4,8 stay within each
        // 16-lane half of the wave32, so each half reduces its own 8 rows.
        #pragma unroll
        for (int off = 8; off >= 1; off >>= 1) {
            #pragma unroll
            for (int j = 0; j < 8; ++j)
                vals[j] += __shfl_xor(vals[j], off, 32);
        }

        // Lane 0 of each half stores its 8 contiguous outputs (32B aligned).
        if (row == 0) {
            float4 o0, o1;
            o0.x = sigmoidf_(vals[0] + bb2);
            o0.y = sigmoidf_(vals[1] + bb2);
            o0.z = sigmoidf_(vals[2] + bb2);
            o0.w = sigmoidf_(vals[3] + bb2);
            o1.x = sigmoidf_(vals[4] + bb2);
            o1.y = sigmoidf_(vals[5] + bb2);
            o1.z = sigmoidf_(vals[6] + bb2);
            o1.w = sigmoidf_(vals[7] + bb2);
            float* op = outB + t * 16 + half * 8;
            *reinterpret_cast<float4*>(op)     = o0;
            *reinterpret_cast<float4*>(op + 4) = o1;
        }
    }
}

extern "C" void kernel_launch(void* const* d_in, const int* in_sizes, int n_in,
                              void* d_out, int out_size, void* d_ws, size_t ws_size,
                              hipStream_t stream) {
    (void)in_sizes; (void)n_in; (void)d_ws; (void)ws_size; (void)out_size;
    const float* atoms = (const float*)d_in[0];
    const int*   gidx  = (const int*)  d_in[1];
    const float* W1    = (const float*)d_in[2];
    const float* b1    = (const float*)d_in[3];
    const float* W2    = (const float*)d_in[4];
    const float* b2    = (const float*)d_in[5];
    float* out = (float*)d_out;

    dim3 grid(CHUNKS, Ff, Bsz);
    cliques_mlp_wmma<<<grid, THREADS, 0, stream>>>(atoms, gidx, W1, b1, W2, b2, out);
}